// SAIDIGraphModel_23570780520828
// MI455X (gfx1250) — compile-verified
//
#include <hip/hip_runtime.h>

typedef float v2f __attribute__((ext_vector_type(2)));
typedef float v8f __attribute__((ext_vector_type(8)));

#define N_NODES 100000
#define N_EDGES 3200000
#define IN_DIM  20
#define HIDDEN  64
#define HALF    32
#define BN_EPS  1e-5f

// ---------------------------------------------------------------------------
// x = concat(xs[ :,6], xt[:,14]) -> [N,20]
// ---------------------------------------------------------------------------
__global__ void concat_kernel(const float* __restrict__ xs,
                              const float* __restrict__ xt,
                              float* __restrict__ x) {
    int i = blockIdx.x * blockDim.x + threadIdx.x;
    if (i >= N_NODES * IN_DIM) return;
    int node = i / IN_DIM, f = i % IN_DIM;
    x[i] = (f < 6) ? xs[node * 6 + f] : xt[node * 14 + (f - 6)];
}

// ---------------------------------------------------------------------------
// deg[dst] += 1 per edge (float degrees, L2-resident atomics)
// ---------------------------------------------------------------------------
__global__ void degree_kernel(const long long* __restrict__ dst,
                              float* __restrict__ deg) {
    int e = blockIdx.x * blockDim.x + threadIdx.x;
    if (e >= N_EDGES) return;
    atomicAdd(&deg[(int)dst[e]], 1.0f);
}

// ---------------------------------------------------------------------------
// Edge-parallel scatter-add of F = F4*4 features per edge.
// float4 gathers (rows are 16B aligned), f32 atomics into L2-resident agg.
// ---------------------------------------------------------------------------
template <int F4>
__global__ void scatter_kernel(const float* __restrict__ feat,
                               const long long* __restrict__ src,
                               const long long* __restrict__ dst,
                               float* __restrict__ agg) {
    const int total = N_EDGES * F4;
    for (int idx = blockIdx.x * blockDim.x + threadIdx.x; idx < total;
         idx += gridDim.x * blockDim.x) {
        int e = idx / F4;
        int q = idx - e * F4;
        int s = (int)src[e];
        int d = (int)dst[e];
        float4 v = ((const float4*)(feat + s * (F4 * 4)))[q];
        float* o = agg + d * (F4 * 4) + q * 4;
        atomicAdd(o + 0, v.x);
        atomicAdd(o + 1, v.y);
        atomicAdd(o + 2, v.z);
        atomicAdd(o + 3, v.w);
    }
}

// ---------------------------------------------------------------------------
// SAGE linear: out = (agg/deg) @ Wl + bl + xroot @ Wr, via V_WMMA_F32_16X16X4_F32.
// One wave per 16x16 output tile; two WMMA accumulations per k-step.
// A layout: lane<16 -> row=m0+lane, K=k,k+1 ; lane>=16 -> row=m0+lane-16, K=k+2,k+3
// B layout: lane<16 -> col=n0+lane, K=k,k+1 ; lane>=16 -> col=n0+lane-16, K=k+2,k+3
// C/D: VGPR r holds (M=r | M=8+r depending on lane half), N = n0 + (lane&15)
// ---------------------------------------------------------------------------
template <int K, int NOUT>
__global__ void sage_gemm_kernel(const float* __restrict__ agg,
                                 const float* __restrict__ deg,
                                 const float* __restrict__ xroot,
                                 const float* __restrict__ Wl,
                                 const float* __restrict__ bl,
                                 const float* __restrict__ Wr,
                                 float* __restrict__ out) {
    constexpr int NTILES = NOUT / 16;
    constexpr int MTILES = N_NODES / 16;   // 6250, exact
    constexpr int KSTEPS = K / 4;

    int wave  = (blockIdx.x * blockDim.x + threadIdx.x) >> 5;
    int lane  = threadIdx.x & 31;
    int mtile = wave / NTILES;
    int ntile = wave - mtile * NTILES;
    if (mtile >= MTILES) return;           // uniform per-wave guard (EXEC stays all-1)

    int laneHi = lane >> 4;                // 0 or 1
    int rowA   = mtile * 16 + (lane & 15);
    int khalf  = laneHi << 1;              // 0 or 2
    int col    = ntile * 16 + (lane & 15);

    float inv = 1.0f / fmaxf(deg[rowA], 1.0f);
    const float* aggRow = agg + rowA * K;
    const float* xRow   = xroot + rowA * K;

    v8f c = {0.f, 0.f, 0.f, 0.f, 0.f, 0.f, 0.f, 0.f};

#pragma unroll
    for (int ks = 0; ks < KSTEPS; ++ks) {
        int k = ks * 4 + khalf;
        v2f a0, b0, a1, b1;
        a0.x = aggRow[k] * inv;  a0.y = aggRow[k + 1] * inv;
        b0.x = Wl[k * NOUT + col];  b0.y = Wl[(k + 1) * NOUT + col];
        c = __builtin_amdgcn_wmma_f32_16x16x4_f32(false, a0, false, b0,
                                                  (short)0, c, false, false);
        a1.x = xRow[k];  a1.y = xRow[k + 1];
        b1.x = Wr[k * NOUT + col];  b1.y = Wr[(k + 1) * NOUT + col];
        c = __builtin_amdgcn_wmma_f32_16x16x4_f32(false, a1, false, b1,
                                                  (short)0, c, false, false);
    }

    float bias = bl[col];
    int rowBase = mtile * 16 + (laneHi << 3);   // +8 for high half-wave
#pragma unroll
    for (int r = 0; r < 8; ++r) {
        out[(rowBase + r) * NOUT + col] = c[r] + bias;
    }
}

// ---------------------------------------------------------------------------
// BN column statistics: LDS-atomic accumulation, one global atomic per column
// per block. sums[0..F) = sum, sums[F..2F) = sum of squares.
// ---------------------------------------------------------------------------
template <int F>
__global__ void bn_stats_kernel(const float* __restrict__ h,
                                float* __restrict__ sums) {
    __shared__ float s_sum[F];
    __shared__ float s_sq[F];
    for (int f = threadIdx.x; f < F; f += blockDim.x) { s_sum[f] = 0.f; s_sq[f] = 0.f; }
    __syncthreads();
    const int total = N_NODES * F;
    for (int idx = blockIdx.x * blockDim.x + threadIdx.x; idx < total;
         idx += gridDim.x * blockDim.x) {
        float v = h[idx];
        int f = idx % F;
        atomicAdd(&s_sum[f], v);
        atomicAdd(&s_sq[f], v * v);
    }
    __syncthreads();
    for (int f = threadIdx.x; f < F; f += blockDim.x) {
        atomicAdd(&sums[f], s_sum[f]);
        atomicAdd(&sums[F + f], s_sq[f]);
    }
}

// ---------------------------------------------------------------------------
// BN (training-mode biased stats) + ReLU
// ---------------------------------------------------------------------------
template <int F>
__global__ void bn_relu_kernel(const float* __restrict__ h,
                               const float* __restrict__ sums,
                               const float* __restrict__ g,
                               const float* __restrict__ be,
                               float* __restrict__ out) {
    int idx = blockIdx.x * blockDim.x + threadIdx.x;
    if (idx >= N_NODES * F) return;
    int f = idx % F;
    float m   = sums[f] * (1.0f / N_NODES);
    float var = sums[F + f] * (1.0f / N_NODES) - m * m;
    float v = (h[idx] - m) * rsqrtf(var + BN_EPS) * g[f] + be[f];
    out[idx] = fmaxf(v, 0.f);
}

// ---------------------------------------------------------------------------
// Fused head: logit = rowmean(h2); embed = relu(logit*Wp+bp); out = embed@Wo+bo
// ---------------------------------------------------------------------------
__global__ void head_kernel(const float* __restrict__ h2,
                            const float* __restrict__ Wp,
                            const float* __restrict__ bp,
                            const float* __restrict__ Wo,
                            const float* __restrict__ bo,
                            float* __restrict__ out) {
    int i = blockIdx.x * blockDim.x + threadIdx.x;
    if (i >= N_NODES) return;
    const float* row = h2 + i * HALF;
    float s = 0.f;
#pragma unroll
    for (int j = 0; j < HALF; ++j) s += row[j];
    float logit = s * (1.0f / HALF);
    float acc = bo[0];
#pragma unroll
    for (int j = 0; j < 32; ++j) {
        float e = fmaxf(logit * Wp[j] + bp[j], 0.f);
        acc += e * Wo[j];
    }
    out[i] = acc;
}

// ---------------------------------------------------------------------------
extern "C" void kernel_launch(void* const* d_in, const int* in_sizes, int n_in,
                              void* d_out, int out_size, void* d_ws, size_t ws_size,
                              hipStream_t stream) {
    const float*     xs   = (const float*)d_in[0];
    const float*     xt   = (const float*)d_in[1];
    const long long* ei   = (const long long*)d_in[2];
    const float*     W1l  = (const float*)d_in[3];
    const float*     b1l  = (const float*)d_in[4];
    const float*     W1r  = (const float*)d_in[5];
    const float*     g1   = (const float*)d_in[6];
    const float*     be1  = (const float*)d_in[7];
    const float*     W2l  = (const float*)d_in[8];
    const float*     b2l  = (const float*)d_in[9];
    const float*     W2r  = (const float*)d_in[10];
    const float*     g2   = (const float*)d_in[11];
    const float*     be2  = (const float*)d_in[12];
    const float*     Wp   = (const float*)d_in[13];
    const float*     bp   = (const float*)d_in[14];
    const float*     Wo   = (const float*)d_in[15];
    const float*     bo   = (const float*)d_in[16];
    float* out = (float*)d_out;

    const long long* src = ei;
    const long long* dst = ei + N_EDGES;

    // Workspace layout (floats), all regions 16B-aligned.
    float* x     = (float*)d_ws;            // N*20 = 2,000,000
    float* deg   = x + N_NODES * IN_DIM;    // N    =   100,000
    float* bufA  = deg + N_NODES;           // N*64 : agg1 (first N*20), then h1
    float* bufB  = bufA + N_NODES * HIDDEN; // N*64 : h1pre, then agg2
    float* bufC  = bufB + N_NODES * HIDDEN; // N*32 : h2pre
    float* bufD  = bufC + N_NODES * HALF;   // N*32 : h2
    float* stats = bufD + N_NODES * HALF;   // 192  : [sum64|sq64|sum32|sq32]

    const int T = 256;

    // Re-zero accumulators every call (ws is not re-poisoned between replays).
    hipMemsetAsync(deg,   0, N_NODES * sizeof(float), stream);
    hipMemsetAsync(bufA,  0, N_NODES * IN_DIM * sizeof(float), stream);
    hipMemsetAsync(stats, 0, 192 * sizeof(float), stream);

    // x = concat(xs, xt)
    concat_kernel<<<(N_NODES * IN_DIM + T - 1) / T, T, 0, stream>>>(xs, xt, x);
    // degrees
    degree_kernel<<<(N_EDGES + T - 1) / T, T, 0, stream>>>(dst, deg);
    // agg1 = scatter-add x[src] (F=20 -> 5 float4 per edge)
    scatter_kernel<5><<<8192, T, 0, stream>>>(x, src, dst, bufA);
    // h1pre = mean@W1l + b1l + x@W1r   (WMMA f32 16x16x4)
    {
        int waves = (N_NODES / 16) * (HIDDEN / 16);     // 25000
        sage_gemm_kernel<IN_DIM, HIDDEN>
            <<<(waves * 32 + T - 1) / T, T, 0, stream>>>(bufA, deg, x, W1l, b1l, W1r, bufB);
    }
    bn_stats_kernel<HIDDEN><<<2048, T, 0, stream>>>(bufB, stats);
    bn_relu_kernel<HIDDEN><<<(N_NODES * HIDDEN + T - 1) / T, T, 0, stream>>>(
        bufB, stats, g1, be1, bufA);                    // h1 -> bufA

    // agg2 = scatter-add h1[src] (F=64 -> 16 float4 per edge); bufB reused
    hipMemsetAsync(bufB, 0, (size_t)N_NODES * HIDDEN * sizeof(float), stream);
    scatter_kernel<16><<<8192, T, 0, stream>>>(bufA, src, dst, bufB);
    // h2pre = mean@W2l + b2l + h1@W2r
    {
        int waves = (N_NODES / 16) * (HALF / 16);       // 12500
        sage_gemm_kernel<HIDDEN, HALF>
            <<<(waves * 32 + T - 1) / T, T, 0, stream>>>(bufB, deg, bufA, W2l, b2l, W2r, bufC);
    }
    bn_stats_kernel<HALF><<<2048, T, 0, stream>>>(bufC, stats + 128);
    bn_relu_kernel<HALF><<<(N_NODES * HALF + T - 1) / T, T, 0, stream>>>(
        bufC, stats + 128, g2, be2, bufD);              // h2 -> bufD

    // heads
    head_kernel<<<(N_NODES + T - 1) / T, T, 0, stream>>>(bufD, Wp, bp, Wo, bo, out);
}